// OnehotGather_35502199668766
// MI455X (gfx1250) — compile-verified
//
#include <hip/hip_runtime.h>
#include <stdint.h>

// Problem geometry (fixed by the reference).
#define GB 8        // batch
#define GS 8192     // sequence length
#define GD 1024     // feature dim (row = 4 KB of fp32)
#define GP 1024     // gathered positions per batch
#define ROW_BYTES (GD * 4)   // 4096

// One 256-thread block (8 wave32) copies one 4 KB row:
//   256 lanes x 16 B = 4096 B.
// Path: global --(global_load_async_to_lds_b128)--> LDS
//              --(global_store_async_from_lds_b128)--> global
// Both ops are CDNA5 async-DMA, tracked by ASYNCcnt; no VGPR staging.
__global__ __launch_bounds__(256) void OnehotGather_gfx1250_kernel(
    const float* __restrict__ seq,   // [B, S, D] fp32
    const int*   __restrict__ pos,   // [B, P] int32 (JAX default x64-off)
    float*       __restrict__ out)   // [B, P, D] fp32
{
    extern __shared__ char smem[];   // 4096 bytes, allocated at launch
    (void)smem;

    const int bp  = blockIdx.x;          // 0 .. B*P-1  (b*P + p)
    const int b   = bp >> 10;            // / GP
    const int tid = threadIdx.x;

    // Row index: uniform across the block (one load, L2/L0-broadcast).
    const int row = pos[bp];

    // Uniform 64-bit bases -> SGPR pairs ("s" constraint forces
    // v_readfirstlane since the compiler can't prove uniformity).
    const uint64_t src_base = (uint64_t)(uintptr_t)seq
        + ((uint64_t)b * GS + (uint64_t)row) * (uint64_t)ROW_BYTES;
    const uint64_t dst_base = (uint64_t)(uintptr_t)out
        + (uint64_t)bp * (uint64_t)ROW_BYTES;

    // Per-lane 32-bit offsets: 16 bytes per lane; LDS offset == chunk offset.
    const uint32_t off = (uint32_t)tid * 16u;   // 0 .. 4080

    // global -> LDS, 128b per lane, async (ASYNCcnt++).
    asm volatile(
        "global_load_async_to_lds_b128 %0, %1, %2\n"
        :
        : "v"(off), "v"(off), "s"(src_base)
        : "memory");

    // Load completion (ASYNCcnt->0) guarantees the LDS writes landed;
    // async LDS accesses are otherwise unordered even within a wave.
    asm volatile("s_wait_asynccnt 0x0" ::: "memory");

    // LDS -> global, 128b per lane, async.
    asm volatile(
        "global_store_async_from_lds_b128 %0, %1, %2\n"
        :
        : "v"(off), "v"(off), "s"(dst_base)
        : "memory");

    // Drain before wave end (S_ENDPGM also implies wait-idle; explicit for clarity).
    asm volatile("s_wait_asynccnt 0x0" ::: "memory");
}

extern "C" void kernel_launch(void* const* d_in, const int* in_sizes, int n_in,
                              void* d_out, int out_size, void* d_ws, size_t ws_size,
                              hipStream_t stream) {
    (void)in_sizes; (void)n_in; (void)out_size; (void)d_ws; (void)ws_size;

    const float* seq = (const float*)d_in[0];   // [8, 8192, 1024] fp32
    const int*   pos = (const int*)d_in[1];     // [8, 1024] int32
    float*       out = (float*)d_out;           // [8, 1024, 1024] fp32

    const int rows = GB * GP;                   // 8192 blocks, one row each
    OnehotGather_gfx1250_kernel<<<rows, 256, ROW_BYTES, stream>>>(seq, pos, out);
}